// GraphConv_90426241450592
// MI455X (gfx1250) — compile-verified
//
#include <hip/hip_runtime.h>
#include <hip/hip_bf16.h>

typedef __attribute__((ext_vector_type(2))) float v2f;
typedef __attribute__((ext_vector_type(8))) float v8f;

#define D 128   // D_IN == D_OUT == 128

// ---------------------------------------------------------------------------
// Kernel 1: fused dual GEMM.
//   out_w0 = verts @ W0 + b0   (written to d_out, initializes accumulation buf)
//   out_w1 = verts @ W1 + b1   (written to d_ws, consumed by scatter kernel)
// Block = 256 threads = 8 waves. Each block owns 64 rows of verts.
// Wave w owns N-tile w (16 cols). Each wave register-blocks 4 M-tiles (64 rows)
// and both weight matrices -> 8 v_wmma_f32_16x16x4_f32 per K-step, 32 K-steps.
// ---------------------------------------------------------------------------
__global__ __launch_bounds__(256) void graphconv_gemm2(
    const float* __restrict__ verts,
    const float* __restrict__ W0, const float* __restrict__ b0,
    const float* __restrict__ W1, const float* __restrict__ b1,
    float* __restrict__ out_w0, float* __restrict__ out_w1, int V)
{
    const int lane   = threadIdx.x & 31;
    const int wave   = threadIdx.x >> 5;   // 0..7 -> N tile
    const int laneLo = lane & 15;
    const int laneHi = lane >> 4;          // 0 or 1 (selects K half / M half)
    const int nBase  = wave * 16;
    const int col    = nBase + laneLo;
    const int rowBase = blockIdx.x * 64;

    v8f acc0[4], acc1[4];
#pragma unroll
    for (int mt = 0; mt < 4; ++mt) { acc0[mt] = (v8f)0.0f; acc1[mt] = (v8f)0.0f; }

    // K loop: 16x16x4 fp32 WMMA, K advances by 4 per step.
#pragma unroll 4
    for (int k = 0; k < D; k += 4) {
        const int kk = k + 2 * laneHi;     // this lane-half holds K = kk, kk+1

        // B fragments (4x16): VGPR0 = rows {K, K+2}, VGPR1 = rows {K+1, K+3}
        v2f bf0, bf1;
        bf0.x = W0[(kk + 0) * D + col];
        bf0.y = W0[(kk + 1) * D + col];
        bf1.x = W1[(kk + 0) * D + col];
        bf1.y = W1[(kk + 1) * D + col];

#pragma unroll
        for (int mt = 0; mt < 4; ++mt) {
            int row = rowBase + mt * 16 + laneLo;
            row = (row < V) ? row : (V - 1);            // clamp tail (stores guarded)
            // A fragment (16x4): lane holds row M=laneLo, K = {kk, kk+1}
            const v2f a = *(const v2f*)(verts + (size_t)row * D + kk);

            acc0[mt] = __builtin_amdgcn_wmma_f32_16x16x4_f32(
                false, a, false, bf0, (short)0, acc0[mt], false, false);
            acc1[mt] = __builtin_amdgcn_wmma_f32_16x16x4_f32(
                false, a, false, bf1, (short)0, acc1[mt], false, false);
        }
    }

    const float bias0 = b0[col];
    const float bias1 = b1[col];

    // D layout: VGPR r -> M = r (lanes 0-15) or M = r+8 (lanes 16-31), N = laneLo
#pragma unroll
    for (int mt = 0; mt < 4; ++mt) {
#pragma unroll
        for (int r = 0; r < 8; ++r) {
            const int row = rowBase + mt * 16 + r + 8 * laneHi;
            if (row < V) {
                out_w0[(size_t)row * D + col] = acc0[mt][r] + bias0;
                out_w1[(size_t)row * D + col] = acc1[mt][r] + bias1;
            }
        }
    }
}

// ---------------------------------------------------------------------------
// Kernel 2: edge scatter.  out[src] += w1v[dst]; out[dst] += w1v[src].
// One wave per edge; lane owns 4 consecutive floats (32 lanes * 4 = 128 cols).
// w1v (51 MB) and out (51 MB) both fit in the 192 MB L2, so gathers + atomics
// are L2-resident. Raw global_atomic_add_f32 via inline asm guarantees the
// hardware float-atomic path (no CAS loop).
// ---------------------------------------------------------------------------
__device__ __forceinline__ void gfadd(float* p, float v) {
    asm volatile("global_atomic_add_f32 %0, %1, off"
                 :: "v"(p), "v"(v) : "memory");
}

__global__ __launch_bounds__(256) void graphconv_scatter(
    const int* __restrict__ edges,
    const float* __restrict__ w1v,
    float* __restrict__ out, int E)
{
    const int e    = (int)((blockIdx.x * blockDim.x + threadIdx.x) >> 5);
    const int lane = threadIdx.x & 31;
    if (e >= E) return;

    const int s = edges[2 * e + 0];
    const int d = edges[2 * e + 1];
    const int c = lane * 4;

    const float4 vd = *(const float4*)(w1v + (size_t)d * D + c);
    const float4 vs = *(const float4*)(w1v + (size_t)s * D + c);

    float* po = out + (size_t)s * D + c;
    gfadd(po + 0, vd.x); gfadd(po + 1, vd.y);
    gfadd(po + 2, vd.z); gfadd(po + 3, vd.w);

    float* pd = out + (size_t)d * D + c;
    gfadd(pd + 0, vs.x); gfadd(pd + 1, vs.y);
    gfadd(pd + 2, vs.z); gfadd(pd + 3, vs.w);
}

// ---------------------------------------------------------------------------
// Launch wrapper
// Inputs (setup_inputs order): verts f32[V*128], edges i32[E*2],
//   W0 f32[128*128], b0 f32[128], W1 f32[128*128], b1 f32[128]
// d_out: f32[V*128]; d_ws: used for verts@W1+b1 (V*128 f32 = 51.2 MB)
// ---------------------------------------------------------------------------
extern "C" void kernel_launch(void* const* d_in, const int* in_sizes, int n_in,
                              void* d_out, int out_size, void* d_ws, size_t ws_size,
                              hipStream_t stream) {
    const float* verts = (const float*)d_in[0];
    const int*   edges = (const int*)d_in[1];
    const float* W0    = (const float*)d_in[2];
    const float* b0    = (const float*)d_in[3];
    const float* W1    = (const float*)d_in[4];
    const float* b1    = (const float*)d_in[5];

    float* out = (float*)d_out;
    float* w1v = (float*)d_ws;

    const int V = in_sizes[0] / D;   // 100000
    const int E = in_sizes[1] / 2;   // 1600000

    // Phase 1: out = verts@W0+b0 (init), w1v = verts@W1+b1
    const int gemmBlocks = (V + 63) / 64;
    graphconv_gemm2<<<gemmBlocks, 256, 0, stream>>>(verts, W0, b0, W1, b1,
                                                    out, w1v, V);

    // Phase 2: scatter over edges (8 edges per 256-thread block)
    const int scatBlocks = (E + 7) / 8;
    graphconv_scatter<<<scatBlocks, 256, 0, stream>>>(edges, w1v, out, E);
}